// GATEdgeConv_31903017075241
// MI455X (gfx1250) — compile-verified
//
#include <hip/hip_runtime.h>

#define HEADS 4
#define CPH 32
#define DIN 128
#define DOUT 128
#define NEG_SLOPE 0.2f
#define EPS_SM 1e-16f
#define EPS_LN 1e-5f

typedef float v2f __attribute__((ext_vector_type(2)));
typedef float v8f __attribute__((ext_vector_type(8)));

__device__ __forceinline__ float leaky(float v) { return v > 0.f ? v : NEG_SLOPE * v; }

// ---------------------------------------------------------------------------
// K1: fused GEMM  [N,128] x [128, 256]  where cols 0..127 -> W (h), 128..255 -> skip_W
// one wave computes a 16-row x 64-col tile via 128 x v_wmma_f32_16x16x4_f32 (32 k-steps x 4 col tiles)
// ---------------------------------------------------------------------------
__global__ __launch_bounds__(32) void k_gemm(const float* __restrict__ x,
                                             const float* __restrict__ W,
                                             const float* __restrict__ skipW,
                                             const float* __restrict__ skip_b,
                                             const float* __restrict__ gat_bias,
                                             float* __restrict__ h,
                                             float* __restrict__ out, int n) {
  const int lane = threadIdx.x;
  const int half = lane >> 4;   // 0: K={0,1}, 1: K={2,3}
  const int lr = lane & 15;
  const int rowBase = blockIdx.x * 16;
  const int cBase = blockIdx.y * 64;  // concat col base 0,64,128,192
  const bool isH = (cBase < DOUT);
  const float* B = isH ? W : skipW;
  const int colBase = isH ? cBase : cBase - DOUT;

  const float* xrow = x + (size_t)(rowBase + lr) * DIN;
  v8f acc[4] = {};
#pragma unroll
  for (int kk = 0; kk < DIN / 4; ++kk) {
    const int kb = kk * 4 + 2 * half;
    // A 16x4 layout: lanes0-15 hold M=lane, K={0,1}; lanes16-31 hold M=lane-16, K={2,3}
    v2f a;
    a.x = xrow[kb];
    a.y = xrow[kb + 1];
#pragma unroll
    for (int t = 0; t < 4; ++t) {
      const int col = colBase + t * 16 + lr;
      // B 4x16 layout: VGPR0 lanes0-15 K=kb+0, lanes16-31 K=kb+2; VGPR1: +1
      v2f b;
      b.x = B[(size_t)kb * DOUT + col];
      b.y = B[(size_t)(kb + 1) * DOUT + col];
      acc[t] = __builtin_amdgcn_wmma_f32_16x16x4_f32(false, a, false, b, (short)0,
                                                     acc[t], false, false);
    }
  }
  // C/D layout: VGPR r -> M = r (lanes0-15) / r+8 (lanes16-31), N = lane&15
#pragma unroll
  for (int t = 0; t < 4; ++t) {
#pragma unroll
    for (int r = 0; r < 8; ++r) {
      const int row = rowBase + r + 8 * half;
      const int col = colBase + t * 16 + lr;
      const float v = acc[t][r];
      if (row < n) {
        if (isH)
          h[(size_t)row * DOUT + col] = v;
        else
          out[(size_t)row * DOUT + col] = v + skip_b[col] + gat_bias[col];
      }
    }
  }
}

// ---------------------------------------------------------------------------
// K2: attention logits a_s[n,h] = <h[n,h,:], att_src[h,:]>, same for a_d
// ---------------------------------------------------------------------------
__global__ void k_attn(const float* __restrict__ h, const float* __restrict__ att_src,
                       const float* __restrict__ att_dst, float* __restrict__ a_s,
                       float* __restrict__ a_d, int n) {
  const int t = blockIdx.x * blockDim.x + threadIdx.x;
  if (t >= n * HEADS) return;
  const int nd = t >> 2, hh = t & 3;
  const float* hr = h + (size_t)nd * DOUT + hh * CPH;
  const float* as = att_src + hh * CPH;
  const float* ad = att_dst + hh * CPH;
  float s0 = 0.f, s1 = 0.f;
#pragma unroll
  for (int c = 0; c < CPH; ++c) {
    const float v = hr[c];
    s0 += v * as[c];
    s1 += v * ad[c];
  }
  a_s[t] = s0;
  a_d[t] = s1;
}

// ---------------------------------------------------------------------------
// CSR build: zero -> histogram -> scan -> scatter (src ids bucketed by dst)
// ---------------------------------------------------------------------------
__global__ void k_zero(unsigned* __restrict__ deg, int n) {
  const int i = blockIdx.x * blockDim.x + threadIdx.x;
  if (i < n) deg[i] = 0u;
}

__global__ void k_hist(const long long* __restrict__ ei, unsigned* __restrict__ deg,
                       int e) {
  const int i = blockIdx.x * blockDim.x + threadIdx.x;
  if (i >= e) return;
  atomicAdd(&deg[(int)ei[(size_t)e + i]], 1u);
}

__global__ __launch_bounds__(1024) void k_scan(const unsigned* __restrict__ deg,
                                               unsigned* __restrict__ off,
                                               unsigned* __restrict__ cursor, int n) {
  __shared__ unsigned sdata[1024];
  __shared__ unsigned srun;
  const int t = threadIdx.x;
  if (t == 0) srun = 0u;
  __syncthreads();
  for (int base = 0; base < n; base += 1024) {
    const int i = base + t;
    const unsigned v = (i < n) ? deg[i] : 0u;
    sdata[t] = v;
    __syncthreads();
    for (int d = 1; d < 1024; d <<= 1) {
      const unsigned add = (t >= d) ? sdata[t - d] : 0u;
      __syncthreads();
      sdata[t] += add;
      __syncthreads();
    }
    const unsigned incl = sdata[t];
    const unsigned excl = incl - v;
    const unsigned run = srun;
    if (i < n) {
      off[i] = run + excl;
      cursor[i] = run + excl;
    }
    __syncthreads();
    if (t == 1023) srun = run + incl;
    __syncthreads();
  }
  if (t == 0) off[n] = srun;
}

__global__ void k_scatter(const long long* __restrict__ ei, unsigned* __restrict__ cursor,
                          unsigned* __restrict__ srcs, int e) {
  const int i = blockIdx.x * blockDim.x + threadIdx.x;
  if (i >= e) return;
  const int d = (int)ei[(size_t)e + i];
  const unsigned pos = atomicAdd(&cursor[d], 1u);
  srcs[pos] = (unsigned)ei[i];
}

// ---------------------------------------------------------------------------
// K7: wave-per-node segment softmax + message aggregation + skip + LayerNorm.
// Self-loop handled inline; no float atomics.
// ---------------------------------------------------------------------------
__global__ __launch_bounds__(256) void k_agg(const float* __restrict__ h,
                                             const float* __restrict__ a_s,
                                             const float* __restrict__ a_d,
                                             const unsigned* __restrict__ off,
                                             const unsigned* __restrict__ srcs,
                                             const float* __restrict__ ln_g,
                                             const float* __restrict__ ln_b,
                                             float* __restrict__ out, int n) {
  const int nid = (int)((blockIdx.x * blockDim.x + threadIdx.x) >> 5);
  const int lane = threadIdx.x & 31;
  if (nid >= n) return;

  const unsigned beg = off[nid], end = off[nid + 1];
  float ad[HEADS], es[HEADS], m[HEADS];
#pragma unroll
  for (int t = 0; t < HEADS; ++t) {
    ad[t] = a_d[nid * HEADS + t];
    es[t] = leaky(a_s[nid * HEADS + t] + ad[t]);  // self-loop logit
    m[t] = es[t];
  }

  // pass 1: segment max (+ prefetch h[src] rows into cache for pass 3)
  for (unsigned i = beg + lane; i < end; i += 32) {
    const unsigned s = srcs[i];
    const float* hp = h + (size_t)s * DOUT;
    __builtin_prefetch(hp, 0, 1);
    __builtin_prefetch(hp + 32, 0, 1);
    __builtin_prefetch(hp + 64, 0, 1);
    __builtin_prefetch(hp + 96, 0, 1);
#pragma unroll
    for (int t = 0; t < HEADS; ++t) {
      const float e = leaky(a_s[s * HEADS + t] + ad[t]);
      m[t] = fmaxf(m[t], e);
    }
  }
#pragma unroll
  for (int t = 0; t < HEADS; ++t) {
    for (int o = 16; o > 0; o >>= 1) m[t] = fmaxf(m[t], __shfl_xor(m[t], o, 32));
  }

  // pass 2: segment sum of exp(e - m)
  float z[HEADS];
#pragma unroll
  for (int t = 0; t < HEADS; ++t) z[t] = (lane == 0) ? expf(es[t] - m[t]) : 0.f;
  for (unsigned i = beg + lane; i < end; i += 32) {
    const unsigned s = srcs[i];
#pragma unroll
    for (int t = 0; t < HEADS; ++t) {
      const float e = leaky(a_s[s * HEADS + t] + ad[t]);
      z[t] += expf(e - m[t]);
    }
  }
  float zinv[HEADS];
#pragma unroll
  for (int t = 0; t < HEADS; ++t) {
    for (int o = 16; o > 0; o >>= 1) z[t] += __shfl_xor(z[t], o, 32);
    zinv[t] = 1.f / (z[t] + EPS_SM);
  }

  // pass 3: weighted gather; lane owns channels [4*lane, 4*lane+3] (one head)
  const int c0 = lane * 4;
  const int hh = lane >> 3;
  const float alpha_self = expf(es[hh] - m[hh]) * zinv[hh];
  const float4 hself = *(const float4*)(h + (size_t)nid * DOUT + c0);
  const float4 basev = *(const float4*)(out + (size_t)nid * DOUT + c0);  // skip+biases
  float acc0 = basev.x + alpha_self * hself.x;
  float acc1 = basev.y + alpha_self * hself.y;
  float acc2 = basev.z + alpha_self * hself.z;
  float acc3 = basev.w + alpha_self * hself.w;
  for (unsigned i = beg; i < end; ++i) {
    const unsigned s = srcs[i];
    const float e = leaky(a_s[s * HEADS + hh] + ad[hh]);
    const float alpha = expf(e - m[hh]) * zinv[hh];
    const float4 hv = *(const float4*)(h + (size_t)s * DOUT + c0);
    acc0 += alpha * hv.x;
    acc1 += alpha * hv.y;
    acc2 += alpha * hv.z;
    acc3 += alpha * hv.w;
  }

  // fused LayerNorm over the 128-wide row
  float sum = acc0 + acc1 + acc2 + acc3;
  float sq = acc0 * acc0 + acc1 * acc1 + acc2 * acc2 + acc3 * acc3;
  for (int o = 16; o > 0; o >>= 1) {
    sum += __shfl_xor(sum, o, 32);
    sq += __shfl_xor(sq, o, 32);
  }
  const float mean = sum * (1.f / DOUT);
  const float var = sq * (1.f / DOUT) - mean * mean;
  const float rstd = rsqrtf(var + EPS_LN);
  const float4 g = *(const float4*)(ln_g + c0);
  const float4 bb = *(const float4*)(ln_b + c0);
  float4 o4;
  o4.x = (acc0 - mean) * rstd * g.x + bb.x;
  o4.y = (acc1 - mean) * rstd * g.y + bb.y;
  o4.z = (acc2 - mean) * rstd * g.z + bb.z;
  o4.w = (acc3 - mean) * rstd * g.w + bb.w;
  *(float4*)(out + (size_t)nid * DOUT + c0) = o4;
}

// ---------------------------------------------------------------------------
extern "C" void kernel_launch(void* const* d_in, const int* in_sizes, int n_in,
                              void* d_out, int out_size, void* d_ws, size_t ws_size,
                              hipStream_t stream) {
  const float* x = (const float*)d_in[0];
  const long long* ei = (const long long*)d_in[1];  // int64 [2,E]
  const float* W = (const float*)d_in[2];
  const float* att_src = (const float*)d_in[3];
  const float* att_dst = (const float*)d_in[4];
  const float* gat_bias = (const float*)d_in[5];
  const float* skipW = (const float*)d_in[6];
  const float* skip_b = (const float*)d_in[7];
  const float* ln_g = (const float*)d_in[8];
  const float* ln_b = (const float*)d_in[9];
  float* out = (float*)d_out;

  const int N = in_sizes[0] / DIN;
  const int E = in_sizes[1] / 2;

  // workspace carve-up (256B aligned slices)
  char* p = (char*)d_ws;
  auto take = [&](size_t bytes) {
    void* r = (void*)p;
    p += (bytes + 255) & ~(size_t)255;
    return r;
  };
  float* h = (float*)take((size_t)N * DOUT * sizeof(float));
  float* a_s = (float*)take((size_t)N * HEADS * sizeof(float));
  float* a_d = (float*)take((size_t)N * HEADS * sizeof(float));
  unsigned* deg = (unsigned*)take((size_t)N * sizeof(unsigned));
  unsigned* cursor = (unsigned*)take((size_t)N * sizeof(unsigned));
  unsigned* off = (unsigned*)take((size_t)(N + 1) * sizeof(unsigned));
  unsigned* srcs = (unsigned*)take((size_t)E * sizeof(unsigned));

  // 1) GEMMs via WMMA: h = x@W ; out = x@skip_W + skip_b + gat_bias
  k_gemm<<<dim3((N + 15) / 16, 4), 32, 0, stream>>>(x, W, skipW, skip_b, gat_bias, h,
                                                    out, N);
  // 2) attention logits
  k_attn<<<(N * HEADS + 255) / 256, 256, 0, stream>>>(h, att_src, att_dst, a_s, a_d, N);
  // 3-6) CSR build by counting sort on dst
  k_zero<<<(N + 255) / 256, 256, 0, stream>>>(deg, N);
  k_hist<<<(E + 255) / 256, 256, 0, stream>>>(ei, deg, E);
  k_scan<<<1, 1024, 0, stream>>>(deg, off, cursor, N);
  k_scatter<<<(E + 255) / 256, 256, 0, stream>>>(ei, cursor, srcs, E);
  // 7) wave-per-node softmax aggregate + skip + LayerNorm (no float atomics)
  k_agg<<<(N + 7) / 8, 256, 0, stream>>>(h, a_s, a_d, off, srcs, ln_g, ln_b, out, N);
}